// GATNet_14628658610611
// MI455X (gfx1250) — compile-verified
//
#include <hip/hip_runtime.h>
#include <math.h>

#define N_NODES 100000
#define E_EDGES 1600000
#define ET_EDGES (E_EDGES + N_NODES)   // with self loops
#define F_IN 128
#define HID 64
#define NEG_SLOPE 0.2f

typedef float v2f __attribute__((ext_vector_type(2)));
typedef float v8f __attribute__((ext_vector_type(8)));

// order-preserving float<->uint mapping for atomicMax on floats
__device__ __forceinline__ unsigned fenc(float f) {
    unsigned b = __float_as_uint(f);
    return (b & 0x80000000u) ? ~b : (b | 0x80000000u);
}
__device__ __forceinline__ float fdec(unsigned e) {
    unsigned b = (e & 0x80000000u) ? (e ^ 0x80000000u) : ~e;
    return __uint_as_float(b);
}

// ---------------- dense: H = X @ W  (W is K x 64), fp32 WMMA ----------------
template <int K>
__global__ void gemm_wmma_f32(const float* __restrict__ X,
                              const float* __restrict__ W,
                              float* __restrict__ H) {
    int wid  = (blockIdx.x * blockDim.x + threadIdx.x) >> 5;  // global wave id
    int lane = threadIdx.x & 31;
    int tm = wid >> 2;      // 16-row tile   (N_NODES/16 = 6250 tiles)
    int tn = wid & 3;       // 16-col tile of 64 output cols
    if (tm >= N_NODES / 16) return;

    int row = tm * 16 + (lane & 15);
    int kk  = (lane >> 4) << 1;            // 0 for lanes 0-15, 2 for lanes 16-31
    int col = tn * 16 + (lane & 15);

    const float* xr = X + (long)row * K;
    v8f acc = {};
#pragma unroll 4
    for (int k0 = 0; k0 < K; k0 += 4) {
        v2f a, b;
        a.x = xr[k0 + kk];
        a.y = xr[k0 + kk + 1];
        b.x = W[(k0 + kk) * HID + col];
        b.y = W[(k0 + kk + 1) * HID + col];
        acc = __builtin_amdgcn_wmma_f32_16x16x4_f32(false, a, false, b,
                                                    (short)0, acc, false, false);
    }
    int rbase = tm * 16 + ((lane >> 4) << 3);  // lanes 16-31 hold rows M=8..15
#pragma unroll
    for (int r = 0; r < 8; ++r)
        H[(long)(rbase + r) * HID + col] = acc[r];
}

// ------------- per-node attention logits: als = h.a_src, ald = h.a_dst ------
__global__ void attn_logits(const float* __restrict__ H,
                            const float* __restrict__ a_src,
                            const float* __restrict__ a_dst,
                            float* __restrict__ als, float* __restrict__ ald) {
    int wid  = (blockIdx.x * blockDim.x + threadIdx.x) >> 5;
    int lane = threadIdx.x & 31;
    if (wid >= N_NODES) return;
    const float* h = H + (long)wid * HID;
    float h0 = h[lane], h1 = h[lane + 32];
    float s = h0 * a_src[lane] + h1 * a_src[lane + 32];
    float d = h0 * a_dst[lane] + h1 * a_dst[lane + 32];
#pragma unroll
    for (int o = 16; o > 0; o >>= 1) {
        s += __shfl_xor(s, o, 32);
        d += __shfl_xor(d, o, 32);
    }
    if (lane == 0) { als[wid] = s; ald[wid] = d; }
}

// ---------------- zero / -inf init of per-layer accumulators ----------------
__global__ void init_bufs(unsigned* __restrict__ m_enc, float* __restrict__ denom,
                          float* __restrict__ num, int numlen) {
    int i = blockIdx.x * blockDim.x + threadIdx.x;
    if (i < N_NODES) { m_enc[i] = 0u; denom[i] = 0.f; }
    if (i < numlen) num[i] = 0.f;
}

// ---------------- edge pass 1: segment max of leaky-relu logits -------------
__global__ void edge_max(const int* __restrict__ ei,
                         const float* __restrict__ als, const float* __restrict__ ald,
                         unsigned* __restrict__ m_enc) {
    int i = blockIdx.x * blockDim.x + threadIdx.x;
    if (i >= ET_EDGES) return;
    int s, d;
    if (i < E_EDGES) { s = ei[i]; d = ei[E_EDGES + i]; }
    else             { s = d = i - E_EDGES; }
    float e = als[s] + ald[d];
    e = e > 0.f ? e : NEG_SLOPE * e;
    atomicMax(m_enc + d, fenc(e));
}

// ---- edge pass 2 (C=64): wave per edge, fused exp / denom / numerator ------
__global__ void edge_acc64(const int* __restrict__ ei,
                           const float* __restrict__ als, const float* __restrict__ ald,
                           const unsigned* __restrict__ m_enc,
                           const float* __restrict__ H,
                           float* __restrict__ num, float* __restrict__ denom) {
    int wid  = (blockIdx.x * blockDim.x + threadIdx.x) >> 5;
    int lane = threadIdx.x & 31;
    if (wid >= ET_EDGES) return;
    int s, d;
    if (wid < E_EDGES) { s = ei[wid]; d = ei[E_EDGES + wid]; }
    else               { s = d = wid - E_EDGES; }
    float e = als[s] + ald[d];
    e = e > 0.f ? e : NEG_SLOPE * e;
    float w = __expf(e - fdec(m_enc[d]));
    if (lane == 0) unsafeAtomicAdd(denom + d, w);
    float h0 = H[(long)s * HID + lane];
    float h1 = H[(long)s * HID + lane + 32];
    unsafeAtomicAdd(num + (long)d * HID + lane,       w * h0);
    unsafeAtomicAdd(num + (long)d * HID + lane + 32,  w * h1);
}

// ---------------- finalize (C=64): divide, bias, optional ELU ---------------
__global__ void finalize64(const float* __restrict__ num, const float* __restrict__ denom,
                           const float* __restrict__ b, float* __restrict__ out, int do_elu) {
    int i = blockIdx.x * blockDim.x + threadIdx.x;
    if (i >= N_NODES * HID) return;
    int n = i >> 6, c = i & 63;
    float v = num[i] / denom[n] + b[c];
    if (do_elu) v = v > 0.f ? v : __expf(v) - 1.f;
    out[i] = v;
}

// ---------------- layer 3 (C=1) helpers -------------------------------------
__global__ void l3_dot(const float* __restrict__ H, const float* __restrict__ W3,
                       const float* __restrict__ as3, const float* __restrict__ ad3,
                       float* __restrict__ h3, float* __restrict__ als,
                       float* __restrict__ ald) {
    int wid  = (blockIdx.x * blockDim.x + threadIdx.x) >> 5;
    int lane = threadIdx.x & 31;
    if (wid >= N_NODES) return;
    float h0 = H[(long)wid * HID + lane];
    float h1 = H[(long)wid * HID + lane + 32];
    float s = h0 * W3[lane] + h1 * W3[lane + 32];
#pragma unroll
    for (int o = 16; o > 0; o >>= 1) s += __shfl_xor(s, o, 32);
    if (lane == 0) {
        h3[wid]  = s;
        als[wid] = s * as3[0];
        ald[wid] = s * ad3[0];
    }
}

__global__ void edge_acc1(const int* __restrict__ ei,
                          const float* __restrict__ als, const float* __restrict__ ald,
                          const unsigned* __restrict__ m_enc,
                          const float* __restrict__ h3,
                          float* __restrict__ num1, float* __restrict__ denom) {
    int i = blockIdx.x * blockDim.x + threadIdx.x;
    if (i >= ET_EDGES) return;
    int s, d;
    if (i < E_EDGES) { s = ei[i]; d = ei[E_EDGES + i]; }
    else             { s = d = i - E_EDGES; }
    float e = als[s] + ald[d];
    e = e > 0.f ? e : NEG_SLOPE * e;
    float w = __expf(e - fdec(m_enc[d]));
    unsafeAtomicAdd(denom + d, w);
    unsafeAtomicAdd(num1 + d, w * h3[s]);
}

__global__ void finalize1(const float* __restrict__ num1, const float* __restrict__ denom,
                          const float* __restrict__ b3, float* __restrict__ out) {
    int i = blockIdx.x * blockDim.x + threadIdx.x;
    if (i >= N_NODES) return;
    out[i] = num1[i] / denom[i] + b3[0];
}

extern "C" void kernel_launch(void* const* d_in, const int* in_sizes, int n_in,
                              void* d_out, int out_size, void* d_ws, size_t ws_size,
                              hipStream_t stream) {
    const float* x      = (const float*)d_in[0];
    const int*   ei     = (const int*)  d_in[1];
    const float* W1     = (const float*)d_in[2];
    const float* a_src1 = (const float*)d_in[3];
    const float* a_dst1 = (const float*)d_in[4];
    const float* b1     = (const float*)d_in[5];
    const float* W2     = (const float*)d_in[6];
    const float* a_src2 = (const float*)d_in[7];
    const float* a_dst2 = (const float*)d_in[8];
    const float* b2     = (const float*)d_in[9];
    const float* W3     = (const float*)d_in[10];
    const float* a_src3 = (const float*)d_in[11];
    const float* a_dst3 = (const float*)d_in[12];
    const float* b3     = (const float*)d_in[13];
    float* out = (float*)d_out;

    // workspace layout (floats)
    float*    bufT  = (float*)d_ws;                 // N*64  transformed features
    float*    bufH  = bufT + (long)N_NODES * HID;   // N*64  layer output/input
    float*    num   = bufH + (long)N_NODES * HID;   // N*64  numerator accumulator
    float*    als   = num  + (long)N_NODES * HID;   // N
    float*    ald   = als  + N_NODES;               // N
    unsigned* m_enc = (unsigned*)(ald + N_NODES);   // N
    float*    denom = (float*)(m_enc + N_NODES);    // N
    float*    h3    = bufT;                         // layer3 reuses bufT
    float*    num1  = num;                          // layer3 reuses num

    const int gemm_blocks  = N_NODES / 16;                 // 4 waves/block, 1 tile each
    const int node_waves   = (N_NODES + 7) / 8;            // 8 waves per 256-thr block
    const int edge_thr_blk = (ET_EDGES + 255) / 256;
    const int edge_wav_blk = (ET_EDGES + 7) / 8;
    const int nc_blocks    = (N_NODES * HID + 255) / 256;
    const int n_blocks     = (N_NODES + 255) / 256;

    // ---------------- layer 1 ----------------
    gemm_wmma_f32<F_IN><<<gemm_blocks, 128, 0, stream>>>(x, W1, bufT);
    attn_logits<<<node_waves, 256, 0, stream>>>(bufT, a_src1, a_dst1, als, ald);
    init_bufs<<<nc_blocks, 256, 0, stream>>>(m_enc, denom, num, N_NODES * HID);
    edge_max<<<edge_thr_blk, 256, 0, stream>>>(ei, als, ald, m_enc);
    edge_acc64<<<edge_wav_blk, 256, 0, stream>>>(ei, als, ald, m_enc, bufT, num, denom);
    finalize64<<<nc_blocks, 256, 0, stream>>>(num, denom, b1, bufH, 1);

    // ---------------- layer 2 ----------------
    gemm_wmma_f32<HID><<<gemm_blocks, 128, 0, stream>>>(bufH, W2, bufT);
    attn_logits<<<node_waves, 256, 0, stream>>>(bufT, a_src2, a_dst2, als, ald);
    init_bufs<<<nc_blocks, 256, 0, stream>>>(m_enc, denom, num, N_NODES * HID);
    edge_max<<<edge_thr_blk, 256, 0, stream>>>(ei, als, ald, m_enc);
    edge_acc64<<<edge_wav_blk, 256, 0, stream>>>(ei, als, ald, m_enc, bufT, num, denom);
    finalize64<<<nc_blocks, 256, 0, stream>>>(num, denom, b2, bufH, 1);

    // ---------------- layer 3 (C = 1) ----------------
    l3_dot<<<node_waves, 256, 0, stream>>>(bufH, W3, a_src3, a_dst3, h3, als, ald);
    init_bufs<<<n_blocks, 256, 0, stream>>>(m_enc, denom, num1, N_NODES);
    edge_max<<<edge_thr_blk, 256, 0, stream>>>(ei, als, ald, m_enc);
    edge_acc1<<<edge_thr_blk, 256, 0, stream>>>(ei, als, ald, m_enc, h3, num1, denom);
    finalize1<<<n_blocks, 256, 0, stream>>>(num1, denom, b3, out);
}